// ChannelWisePatchLevelObfuscator_67860483277431
// MI455X (gfx1250) — compile-verified
//
#include <hip/hip_runtime.h>
#include <math.h>

typedef float v2f __attribute__((ext_vector_type(2)));
typedef float v8f __attribute__((ext_vector_type(8)));

#define IMG 512
#define NCH 3
#define PS  16
#define NG  32
#define NB  64
#define NW_ 32
#define NH_ 32
#define P2  256
#define KC  32
#define NKC 8     // 256 / KC
#define LDA 36    // A row stride (dwords): conflict-free b64 reads, 16B-aligned rows
#define LDT 36    // transposed-W row stride (dwords): conflict-free b64 reads

// ---- CDNA5 async global->LDS copies (ASYNCcnt-tracked, no VGPR roundtrip) ----
__device__ __forceinline__ void async_cp_b128(unsigned lds_off, const void* gptr) {
    asm volatile("global_load_async_to_lds_b128 %0, %1, off"
                 :: "v"(lds_off), "v"(gptr) : "memory");
}
__device__ __forceinline__ void async_cp_b32(unsigned lds_off, const void* gptr) {
    asm volatile("global_load_async_to_lds_b32 %0, %1, off"
                 :: "v"(lds_off), "v"(gptr) : "memory");
}
__device__ __forceinline__ void async_wait0() {
#if __has_builtin(__builtin_amdgcn_s_wait_asynccnt)
    __builtin_amdgcn_s_wait_asynccnt(0);
#else
    asm volatile("s_wait_asynccnt 0x0" ::: "memory");
#endif
}
// low 32 bits of a flat pointer into LDS == wave-relative LDS byte offset
__device__ __forceinline__ unsigned lds_off(const void* p) {
    return (unsigned)(unsigned long long)p;
}

__launch_bounds__(256)
__global__ void obf_wmma_kernel(const float* __restrict__ x,
                                const float* __restrict__ w,
                                const float* __restrict__ bias,
                                const int*   __restrict__ perm,
                                float* __restrict__ out)
{
    __shared__ float lA [2][NB * LDA];   // 2 x 64 x 36   (18 KB)
    __shared__ float lWt[2][P2 * LDT];   // 2 x 256 x 36  (73.7 KB) — W transposed

    const int bid = blockIdx.x;
    const int c   = bid / (NH_ * NW_);
    const int n   = bid % (NH_ * NW_);
    const int nh  = n / NW_;
    const int nw  = n % NW_;
    const int g   = (nh + nw) % NG;

    const int tid  = threadIdx.x;
    const int lane = tid & 31;
    const int wv   = tid >> 5;            // wave id 0..7

    // inverse channel permutation: out channel j takes input channel perm[j]
    int cdst = 0;
    #pragma unroll
    for (int j = 0; j < NCH; ++j) if (perm[j] == c) cdst = j;

    const float* wg = w    + (size_t)(c * NG + g) * P2 * P2;  // 256x256 row-major
    const float* bg = bias + (size_t)(c * NG + g) * P2;

    const int m0    = (wv & 3) * 16;      // M-tile origin: 0,16,32,48
    const int nbase = (wv >> 2) * 128;    // N-half origin: 0 or 128

    v8f acc[8];
    const v8f vzero = {0.f,0.f,0.f,0.f,0.f,0.f,0.f,0.f};
    #pragma unroll
    for (int j = 0; j < 8; ++j) acc[j] = vzero;

    const int hh = lane >> 4;             // half-wave (0/1)
    const int ll = lane & 15;

    // ---- async staging of one K-chunk into buffer `buf` ----
    // A: rows b=0..63, 32 floats (2 patch-rows) each; thread -> (row, 8-float part)
    const int arow  = tid >> 2;
    const int apro  = (tid >> 1) & 1;     // which of the 2 patch-rows
    const int ahalf = tid & 1;            // which 8-float half
    // W (transposed): lanes sweep columns (coalesced global), LDS scattered by engine
    auto stage = [&](int kc, int buf) {
        const int k0 = kc * KC;
        // A chunk
        {
            const int pr = kc * 2 + apro;
            const float* src = x + ((size_t)(arow * NCH + c) * IMG + nh * PS + pr) * IMG
                                 + nw * PS + ahalf * 8;
            const unsigned dst = lds_off(&lA[buf][arow * LDA + apro * 16 + ahalf * 8]);
            async_cp_b128(dst,       src);
            async_cp_b128(dst + 16u, src + 4);
        }
        // W chunk, transposed into LDS
        #pragma unroll
        for (int it = 0; it < 2; ++it) {
            #pragma unroll
            for (int r2 = 0; r2 < 2; ++r2) {
                const int row = (it * 8 + wv) * 2 + r2;          // 0..31
                const float* srow = wg + (size_t)(k0 + row) * P2;
                #pragma unroll
                for (int ob = 0; ob < 8; ++ob) {
                    const int o = ob * 32 + lane;                // coalesced in global
                    async_cp_b32(lds_off(&lWt[buf][o * LDT + row]), srow + o);
                }
            }
        }
    };

    stage(0, 0);
    async_wait0();
    __syncthreads();

    for (int kc = 0; kc < NKC; ++kc) {
        const int buf = kc & 1;

        // overlap: async engine fills the other buffer while we do WMMAs
        if (kc < NKC - 1) stage(kc + 1, buf ^ 1);
        if (kc < NKC - 2)   // warm L2 for the chunk after that
            __builtin_prefetch(wg + (size_t)((kc + 2) * KC + (tid >> 3)) * P2
                                  + (tid & 7) * 32, 0, 1);

        // ---- compute: K steps of 4 via V_WMMA_F32_16X16X4_F32 ----
        #pragma unroll
        for (int kk = 0; kk < KC; kk += 4) {
            // A fragment 16x4: lane<16 -> (K=kk,kk+1); lane>=16 -> (kk+2,kk+3)
            v2f a = *(const v2f*)&lA[buf][(m0 + ll) * LDA + kk + 2 * hh];
            #pragma unroll
            for (int j = 0; j < 8; ++j) {
                const int n0 = nbase + j * 16;
                // B fragment: transposed W -> single aligned b64 (rows k,k+1 @ col n0+ll)
                v2f bf = *(const v2f*)&lWt[buf][(n0 + ll) * LDT + kk + 2 * hh];
                acc[j] = __builtin_amdgcn_wmma_f32_16x16x4_f32(
                    false, a, false, bf, (short)0, acc[j], false, false);
            }
        }

        async_wait0();        // own next-chunk copies complete
        __syncthreads();      // all waves done computing `buf` / loading `buf^1`
    }

    // ---- epilogue: bias + tanh, scatter to permuted output channel ----
    #pragma unroll
    for (int j = 0; j < 8; ++j) {
        const int n0   = nbase + j * 16;
        const int o    = n0 + ll;         // output-pixel index within patch
        const int prow = n0 >> 4;         // o / 16 (ll < 16)
        const float bv = bg[o];
        #pragma unroll
        for (int r = 0; r < 8; ++r) {
            const int brow = m0 + r + 8 * hh;   // C/D layout: M = r + 8*half
            float v = tanhf(acc[j][r] + bv);
            out[((size_t)(brow * NCH + cdst) * IMG + nh * PS + prow) * IMG
                + nw * PS + ll] = v;
        }
    }
}

extern "C" void kernel_launch(void* const* d_in, const int* in_sizes, int n_in,
                              void* d_out, int out_size, void* d_ws, size_t ws_size,
                              hipStream_t stream) {
    const float* x    = (const float*)d_in[0];
    const float* w    = (const float*)d_in[1];
    const float* bias = (const float*)d_in[2];
    const int*   perm = (const int*)  d_in[3];
    float*       out  = (float*)d_out;

    dim3 grid(NCH * NH_ * NW_);   // 3072 workgroups: one per (channel, patch)
    dim3 block(256);              // 8 wave32
    obf_wmma_kernel<<<grid, block, 0, stream>>>(x, w, bias, perm, out);
}